// Attention_64819646431797
// MI455X (gfx1250) — compile-verified
//
#include <hip/hip_runtime.h>
#include <math.h>

#define B_    32
#define H_    32
#define KVH_  8
#define G_    4
#define D_    128
#define BS_   16
#define BPS_  129               // BLOCKS_PER_SEQ
#define QROW_ 132               // qbuf row stride (floats): 528B, 16B aligned, bank stride 4
#define HROW_ 68                // K half-tile row stride (floats): 272B, 16B aligned, bank stride 4
#define SCALE_ 0.08838834764831845f   // 1/sqrt(128)

typedef float v2f __attribute__((ext_vector_type(2)));
typedef float v8f __attribute__((ext_vector_type(8)));

__device__ __forceinline__ void lds_fence() {
    asm volatile("s_wait_dscnt 0x0" ::: "memory");
}
__device__ __forceinline__ void wait_async_le8() {
    asm volatile("s_wait_asynccnt 0x8" ::: "memory");
}
__device__ __forceinline__ void wait_async_0() {
    asm volatile("s_wait_asynccnt 0x0" ::: "memory");
}
// Async DMA: global -> LDS, 16B per lane, no VGPR data round-trip (ASYNCcnt-tracked)
__device__ __forceinline__ void async_ld_b128(unsigned lds_off, const float* g) {
    asm volatile("global_load_async_to_lds_b128 %0, %1, off"
                 :: "v"(lds_off), "v"(g) : "memory");
}

// Stage one half-tile (16 keys x 64 dims): 8 async b128 instructions,
// each moving two contiguous 256B key half-rows (lanes 0-15 / 16-31).
__device__ __forceinline__ void stage_half(const float* __restrict__ kc, int blk, int kvh,
                                           int h, float* buf, int lane) {
    const int keyhi = lane >> 4;          // 0 or 1
    const int col   = (lane & 15) * 4;    // float offset within half-row
    #pragma unroll
    for (int i = 0; i < 8; ++i) {
        const int kk = i * 2 + keyhi;
        const float* g = kc + ((size_t)(blk * BS_ + kk) * KVH_ + kvh) * D_ + h * 64 + col;
        unsigned l = (unsigned)(unsigned long long)(buf + kk * HROW_ + col);
        async_ld_b128(l, g);
    }
}

__device__ __forceinline__ v8f score_half(v8f cs, const float* qb, const float* kbh,
                                          int h, int ln16, int rhalf) {
    #pragma unroll
    for (int cc = 0; cc < 16; ++cc) {
        int off = cc * 4 + rhalf;
        v2f a  = *(const v2f*)(qb  + ln16 * QROW_ + h * 64 + off);
        v2f bf = *(const v2f*)(kbh + ln16 * HROW_ + off);
        cs = __builtin_amdgcn_wmma_f32_16x16x4_f32(
                 false, a, false, bf, (short)0, cs, false, false);
    }
    return cs;
}

__launch_bounds__(128)
__global__ void pa_decode_kernel(const float* __restrict__ q,
                                 const float* __restrict__ k,
                                 const float* __restrict__ v,
                                 const float* __restrict__ kc,
                                 const float* __restrict__ vc,
                                 const int*   __restrict__ bt,
                                 const int*   __restrict__ cl,
                                 float*       __restrict__ out)
{
    const int bh   = blockIdx.x;
    const int b    = bh / KVH_;
    const int kvh  = bh % KVH_;
    const int tid  = threadIdx.x;
    const int w    = tid >> 5;          // wave id 0..3
    const int lane = tid & 31;
    const int ln16 = lane & 15;
    const int rhalf = (lane >> 4) << 1; // 0 for lanes 0-15, 2 for lanes 16-31

    __shared__ __align__(16) float qbuf[16 * QROW_];        // Q tile, rows 4..15 = 0
    __shared__ __align__(16) float kbufs[4][2][16 * HROW_]; // per-wave double-buffered K half-tiles
    __shared__ __align__(16) float wbuf[4][256];            // per-wave weights (16x16, rows 4..15 = 0)
    __shared__ __align__(16) float pacc[16][D_];            // per (wave,row) partial output
    __shared__ float pm[16], pl[16], scur[4];

    // ---- stage Q (zero-padded to 16 rows) ----
    for (int i = tid; i < 16 * QROW_; i += 128) qbuf[i] = 0.f;
    for (int i = lane; i < 256; i += 32) wbuf[w][i] = 0.f;
    __syncthreads();
    for (int i = tid; i < G_ * D_; i += 128) {
        int row = i >> 7, d = i & (D_ - 1);
        qbuf[row * QROW_ + d] = q[((size_t)(b * H_ + kvh * G_ + row)) * D_ + d];
    }
    __syncthreads();

    const int ctx    = cl[b];
    const int ntiles = (ctx + BS_ - 1) / BS_;

    v8f   acc[8] = {};
    float mrow[4] = {-INFINITY, -INFINITY, -INFINITY, -INFINITY};
    float lrow[4] = {0.f, 0.f, 0.f, 0.f};

    float* kb0 = kbufs[w][0];
    float* kb1 = kbufs[w][1];
    float* wb  = wbuf[w];

    // ---- prologue: put both half-tiles of this wave's first tile in flight ----
    if (w < ntiles) {
        const int blk0 = bt[b * BPS_ + w];
        stage_half(kc, blk0, kvh, 0, kb0, lane);
        stage_half(kc, blk0, kvh, 1, kb1, lane);
    }

    for (int tile = w; tile < ntiles; tile += 4) {
        const int t0  = tile * BS_;
        const int blk = bt[b * BPS_ + tile];
        const bool has_next = (tile + 4) < ntiles;
        const int nblk = has_next ? bt[b * BPS_ + tile + 4] : 0;

        if (has_next) {   // warm L2 for next tile's V (demand-loaded later)
            const char* pv = (const char*)(vc + ((size_t)(nblk * BS_ + ln16) * KVH_ + kvh) * D_)
                             + (lane >> 4) * 256;
            __builtin_prefetch(pv, 0, 0);
        }

        // ---- scores: C(16x16) = Q(16x4) x K^T(4x16), 32 d-chunks via 2 async half-tiles ----
        v8f cs = {};
        wait_async_le8();                         // half 0 landed (oldest 8 of <=16)
        cs = score_half(cs, qbuf, kb0, 0, ln16, rhalf);
        lds_fence();                              // frag reads retired before DMA overwrite
        if (has_next) stage_half(kc, nblk, kvh, 0, kb0, lane);

        if (has_next) wait_async_le8(); else wait_async_0();   // half 1 landed
        cs = score_half(cs, qbuf, kb1, 1, ln16, rhalf);
        lds_fence();
        if (has_next) stage_half(kc, nblk, kvh, 1, kb1, lane);

        // ---- online softmax (rows 0..3 live in lanes 0-15, VGPR j) ----
        float sc[4];
        #pragma unroll
        for (int j = 0; j < 4; ++j) {
            float s = cs[j] * SCALE_;
            if (t0 + ln16 >= ctx) s = -INFINITY;
            sc[j] = s;
        }
        #pragma unroll
        for (int j = 0; j < 4; ++j) {
            float m = sc[j];
            #pragma unroll
            for (int off = 8; off; off >>= 1) m = fmaxf(m, __shfl_xor(m, off, 16));
            float mx = fmaxf(mrow[j], m);
            float al = __expf(mrow[j] - mx);
            float wv = __expf(sc[j] - mx);
            float rs = wv;
            #pragma unroll
            for (int off = 8; off; off >>= 1) rs += __shfl_xor(rs, off, 16);
            lrow[j] = lrow[j] * al + rs;
            mrow[j] = mx;
            #pragma unroll
            for (int nt = 0; nt < 8; ++nt) acc[nt][j] *= al;
            if (lane < 16) wb[j * 16 + ln16] = wv;
        }
        lds_fence();

        // ---- PV: acc += W(16x4) x V(4x16) per 16-dim d-tile; V coalesced from HBM ----
        #pragma unroll
        for (int c4 = 0; c4 < 4; ++c4) {
            v2f a = *(const v2f*)(wb + ln16 * 16 + c4 * 4 + rhalf);
            const int key0 = c4 * 4 + rhalf;
            const float* v0 = vc + ((size_t)(blk * BS_ + key0) * KVH_ + kvh) * D_;
            const float* v1 = v0 + (size_t)KVH_ * D_;
            #pragma unroll
            for (int nt = 0; nt < 8; ++nt) {
                v2f bf;
                bf.x = v0[nt * 16 + ln16];
                bf.y = v1[nt * 16 + ln16];
                acc[nt] = __builtin_amdgcn_wmma_f32_16x16x4_f32(
                              false, a, false, bf, (short)0, acc[nt], false, false);
            }
        }
    }

    // ---- current (appended) token: wave w handles q-head j = w ----
    {
        const float* qrow = q + ((size_t)(b * H_ + kvh * G_ + w)) * D_;
        const float* krow = k + ((size_t)(b * KVH_ + kvh)) * D_;
        float4 qa = *(const float4*)(qrow + lane * 4);
        float4 ka = *(const float4*)(krow + lane * 4);
        float p = qa.x * ka.x + qa.y * ka.y + qa.z * ka.z + qa.w * ka.w;
        #pragma unroll
        for (int off = 16; off; off >>= 1) p += __shfl_xor(p, off, 32);
        if (lane == 0) scur[w] = p * SCALE_;
    }

    // ---- publish per-wave partials ----
    if (lane < 16) {
        #pragma unroll
        for (int j = 0; j < 4; ++j) {
            #pragma unroll
            for (int nt = 0; nt < 8; ++nt)
                pacc[w * 4 + j][nt * 16 + ln16] = acc[nt][j];
        }
        if (lane == 0) {
            #pragma unroll
            for (int j = 0; j < 4; ++j) { pm[w * 4 + j] = mrow[j]; pl[w * 4 + j] = lrow[j]; }
        }
    }
    __syncthreads();

    // ---- merge 4 wave partials + current token; thread t owns dim d = t ----
    const int d = tid;
    const float vcur = v[((size_t)(b * KVH_ + kvh)) * D_ + d];
    #pragma unroll
    for (int j = 0; j < 4; ++j) {
        float s0 = scur[j];
        float M2 = s0;
        #pragma unroll
        for (int ww = 0; ww < 4; ++ww) M2 = fmaxf(M2, pm[ww * 4 + j]);
        float e0  = __expf(s0 - M2);
        float den = e0;
        float num = e0 * vcur;
        #pragma unroll
        for (int ww = 0; ww < 4; ++ww) {
            float e = __expf(pm[ww * 4 + j] - M2);
            den += pl[ww * 4 + j] * e;
            num += pacc[ww * 4 + j][d] * e;
        }
        out[((size_t)(b * H_ + kvh * G_ + j)) * D_ + d] = num / den;
    }
}

extern "C" void kernel_launch(void* const* d_in, const int* in_sizes, int n_in,
                              void* d_out, int out_size, void* d_ws, size_t ws_size,
                              hipStream_t stream) {
    (void)in_sizes; (void)n_in; (void)d_ws; (void)ws_size; (void)out_size;
    const float* q  = (const float*)d_in[0];
    const float* k  = (const float*)d_in[1];
    const float* v  = (const float*)d_in[2];
    const float* kc = (const float*)d_in[3];
    const float* vc = (const float*)d_in[4];
    const int*   bt = (const int*)d_in[5];
    const int*   cl = (const int*)d_in[6];
    // d_in[7] = block_size scalar (compile-time 16 here)
    pa_decode_kernel<<<dim3(B_ * KVH_), dim3(128), 0, stream>>>(
        q, k, v, kc, vc, bt, cl, (float*)d_out);
}